// ODENet_83399674954173
// MI455X (gfx1250) — compile-verified
//
#include <hip/hip_runtime.h>
#include <hip/hip_bf16.h>

typedef float v2f __attribute__((ext_vector_type(2)));
typedef float v8f __attribute__((ext_vector_type(8)));

#define NB      131072
#define DIM     128
#define NSTEP   32
#define FUTN    16
#define LN_EPS  1e-5f
#define PITCH   132   // LDS row pitch in floats (bank-conflict avoidance)

// ---------------- common device helpers ----------------

__device__ __forceinline__ float tanh_fast(float x) {
    // tanh(x) = 1 - 2/(e^{2x}+1); robust at +-inf
    float e = __expf(2.0f * x);
    return 1.0f - 2.0f / (e + 1.0f);
}

// ---------------- wave-level WMMA tile machinery ----------------
// Layouts (wave32, V_WMMA_F32_16X16X4_F32):
//  C/D (16x16 f32): element (m,n) -> lane = (n&15) + 16*(m/8), vgpr = m%8
//  A   (16x4  f32): element (m,k) -> lane = m + 16*((k&3)/2), vgpr = k%2
//  B   (4x16  f32): element (k,n) -> lane = n + 16*((k&3)/2), vgpr = k%2
// Weight matrices are pre-shuffled into "fragment order":
//  float2 at Wf[ ((kc*8 + t)*32 + lane)*2 ] holds B[kc*4+2*(lane/16)+j][t*16+lane%16]

__device__ __forceinline__ void wv_gemm(const float* __restrict__ Xs,
                                        const float* __restrict__ Wf,
                                        v8f acc[8], int lane) {
#pragma unroll
    for (int t = 0; t < 8; ++t) {
#pragma unroll
        for (int p = 0; p < 8; ++p) acc[t][p] = 0.0f;
    }
    const int l16 = lane & 15, half = lane >> 4;
    const float* ar = Xs + l16 * PITCH + 2 * half;
    const float* wb = Wf + lane * 2;
#pragma unroll 4
    for (int kc = 0; kc < 32; ++kc) {
        v2f a = *(const v2f*)(ar + kc * 4);
        const float* wk = wb + kc * 512;
#pragma unroll
        for (int t = 0; t < 8; ++t) {
            v2f b = *(const v2f*)(wk + t * 64);
            acc[t] = __builtin_amdgcn_wmma_f32_16x16x4_f32(
                false, a, false, b, (short)0, acc[t], false, false);
        }
    }
}

__device__ __forceinline__ void wv_store(float* Xs, const v8f a[8], int lane) {
    const int l16 = lane & 15, half = lane >> 4;
#pragma unroll
    for (int t = 0; t < 8; ++t)
#pragma unroll
        for (int p = 0; p < 8; ++p)
            Xs[(p + 8 * half) * PITCH + t * 16 + l16] = a[t][p];
}

__device__ __forceinline__ void wv_store_axpy(float* Xs, const v8f y[8],
                                              const v8f k[8], float c, int lane) {
    const int l16 = lane & 15, half = lane >> 4;
#pragma unroll
    for (int t = 0; t < 8; ++t)
#pragma unroll
        for (int p = 0; p < 8; ++p)
            Xs[(p + 8 * half) * PITCH + t * 16 + l16] = y[t][p] + c * k[t][p];
}

__device__ __forceinline__ void wv_bias_relu(v8f a[8], const float* __restrict__ bias,
                                             int lane) {
    const int l16 = lane & 15;
#pragma unroll
    for (int t = 0; t < 8; ++t) {
        float bv = bias[t * 16 + l16];
#pragma unroll
        for (int p = 0; p < 8; ++p) {
            float v = a[t][p] + bv;
            a[t][p] = v > 0.0f ? v : 0.0f;
        }
    }
}

__device__ __forceinline__ void wv_bias_tanh(v8f a[8], const float* __restrict__ bias,
                                             int lane) {
    const int l16 = lane & 15;
#pragma unroll
    for (int t = 0; t < 8; ++t) {
        float bv = bias[t * 16 + l16];
#pragma unroll
        for (int p = 0; p < 8; ++p) a[t][p] = tanh_fast(a[t][p] + bv);
    }
}

// butterfly-reduce 8 per-row partials across each 16-lane group
__device__ __forceinline__ void wv_rowred(float s[8]) {
#pragma unroll
    for (int m = 1; m < 16; m <<= 1) {
#pragma unroll
        for (int p = 0; p < 8; ++p) s[p] += __shfl_xor(s[p], m, 32);
    }
}

__device__ __forceinline__ void wv_rowstats(const v8f a[8], float mean[8], float inv[8]) {
    float s[8];
#pragma unroll
    for (int p = 0; p < 8; ++p) s[p] = 0.0f;
#pragma unroll
    for (int t = 0; t < 8; ++t)
#pragma unroll
        for (int p = 0; p < 8; ++p) s[p] += a[t][p];
    wv_rowred(s);
#pragma unroll
    for (int p = 0; p < 8; ++p) { mean[p] = s[p] * (1.0f / 128.0f); s[p] = 0.0f; }
#pragma unroll
    for (int t = 0; t < 8; ++t)
#pragma unroll
        for (int p = 0; p < 8; ++p) {
            float d = a[t][p] - mean[p];
            s[p] += d * d;
        }
    wv_rowred(s);
#pragma unroll
    for (int p = 0; p < 8; ++p) inv[p] = rsqrtf(s[p] * (1.0f / 128.0f) + LN_EPS);
}

__device__ __forceinline__ void wv_layernorm(v8f a[8], const float* __restrict__ g,
                                             const float* __restrict__ be, int lane) {
    const int l16 = lane & 15;
    float mean[8], inv[8];
    wv_rowstats(a, mean, inv);
#pragma unroll
    for (int t = 0; t < 8; ++t) {
        float gv = g[t * 16 + l16], bv = be[t * 16 + l16];
#pragma unroll
        for (int p = 0; p < 8; ++p)
            a[t][p] = (a[t][p] - mean[p]) * inv[p] * gv + bv;
    }
}

// o[p] = LN(a; g,be) . w + b2s (non-destructive)
__device__ __forceinline__ void wv_ln_dot(const v8f a[8], const float* __restrict__ g,
                                          const float* __restrict__ be,
                                          const float* __restrict__ w, float b2s,
                                          float o[8], int lane) {
    const int l16 = lane & 15;
    float mean[8], inv[8];
    wv_rowstats(a, mean, inv);
    float d[8];
#pragma unroll
    for (int p = 0; p < 8; ++p) d[p] = 0.0f;
#pragma unroll
    for (int t = 0; t < 8; ++t) {
        float gv = g[t * 16 + l16], bv = be[t * 16 + l16], wvv = w[t * 16 + l16];
#pragma unroll
        for (int p = 0; p < 8; ++p)
            d[p] += ((a[t][p] - mean[p]) * inv[p] * gv + bv) * wvv;
    }
    wv_rowred(d);
#pragma unroll
    for (int p = 0; p < 8; ++p) o[p] = d[p] + b2s;
}

// fdyn: input tile already staged in Xs; output in 'out'
__device__ __forceinline__ void wv_fdyn(float* Xs, v8f out[8],
                                        const float* __restrict__ A1f,
                                        const float* __restrict__ A2f,
                                        const float* __restrict__ A3f,
                                        const float* __restrict__ a1p,
                                        const float* __restrict__ a2p,
                                        const float* __restrict__ a3p, int lane) {
    wv_gemm(Xs, A1f, out, lane);
    wv_bias_relu(out, a1p, lane);
    wv_store(Xs, out, lane);
    wv_gemm(Xs, A2f, out, lane);
    wv_bias_relu(out, a2p, lane);
    wv_store(Xs, out, lane);
    wv_gemm(Xs, A3f, out, lane);
    wv_bias_tanh(out, a3p, lane);
}

// ---------------- kernel 0: weight prep (pad + fragment reorder) ----------------
// ws layout (floats): W1f@0, A1f@16384, A2f@32768, A3f@49152,
//                     b1p@65536, a1p@65664, a2p@65792, a3p@65920  (total 66048)

__global__ void ode_prep_kernel(const float* __restrict__ W1, const float* __restrict__ A1,
                                const float* __restrict__ A2, const float* __restrict__ A3,
                                const float* __restrict__ b1, const float* __restrict__ a1,
                                const float* __restrict__ a2, const float* __restrict__ a3,
                                float* __restrict__ ws) {
    int id = blockIdx.x * blockDim.x + threadIdx.x;
    if (id < 4 * 16384) {
        int m = id >> 14;
        int r = id & 16383;
        int j = r & 1;
        int lane = (r >> 1) & 31;
        int f = r >> 6;
        int t = f & 7;
        int kc = f >> 3;
        int k = kc * 4 + 2 * (lane >> 4) + j;
        int n = t * 16 + (lane & 15);
        const float* src; int K, N;
        if (m == 0)      { src = W1; K = 128; N = 128; }
        else if (m == 1) { src = A1; K = 128; N = 124; }
        else if (m == 2) { src = A2; K = 124; N = 124; }
        else             { src = A3; K = 124; N = 128; }
        float v = (k < K && n < N) ? src[k * N + n] : 0.0f;
        ws[m * 16384 + r] = v;
    } else if (id < 4 * 16384 + 512) {
        int r = id - 65536;
        int m = r >> 7;
        int n = r & 127;
        float v;
        if (m == 0)      v = b1[n];
        else if (m == 1) v = (n < 124) ? a1[n] : 0.0f;
        else if (m == 2) v = (n < 124) ? a2[n] : 0.0f;
        else             v = a3[n];
        ws[65536 + r] = v;
    }
}

// ---------------- kernel 1: big-batch pipeline (WMMA) ----------------

__global__ __launch_bounds__(128) void ode_batch_kernel(
    const float* __restrict__ x,
    const float* __restrict__ g1, const float* __restrict__ beta1,
    const float* __restrict__ g2, const float* __restrict__ beta2,
    const float* __restrict__ W2, const float* __restrict__ b2,
    const float* __restrict__ W1f, const float* __restrict__ A1f,
    const float* __restrict__ A2f, const float* __restrict__ A3f,
    const float* __restrict__ b1p, const float* __restrict__ a1p,
    const float* __restrict__ a2p, const float* __restrict__ a3p,
    float* __restrict__ out) {
    __shared__ __align__(16) float lds[4 * 16 * PITCH];
    const int lane = threadIdx.x & 31;
    const int wave = threadIdx.x >> 5;
    float* Xs = lds + wave * (16 * PITCH);
    const int tile = blockIdx.x * 4 + wave;
    const int base = tile * 16;

    // load 16x128 tile of x into padded LDS (coalesced float4)
#pragma unroll
    for (int i = 0; i < 16; ++i) {
        int vid = i * 32 + lane;           // 512 float4 per tile
        int row = vid >> 5;                // 32 float4 per row
        int c4 = (vid & 31) * 4;
        float4 v = *(const float4*)(x + (size_t)(base + row) * DIM + c4);
        *(float4*)(Xs + row * PITCH + c4) = v;
    }

    v8f h[8];
    wv_gemm(Xs, W1f, h, lane);
    wv_bias_relu(h, b1p, lane);
    wv_layernorm(h, g1, beta1, lane);

    const float b2s = b2[0];
    float oh[8];
    wv_ln_dot(h, g2, beta2, W2, b2s, oh, lane);

    v8f y[8], k[8], sum[8];
#pragma unroll
    for (int t = 0; t < 8; ++t) y[t] = h[t];

    const float dt = 1.0f / (float)NSTEP;
    for (int s = 0; s < NSTEP; ++s) {
        wv_store(Xs, y, lane);
        wv_fdyn(Xs, k, A1f, A2f, A3f, a1p, a2p, a3p, lane);            // k1
#pragma unroll
        for (int t = 0; t < 8; ++t) sum[t] = k[t];
        wv_store_axpy(Xs, y, k, 0.5f * dt, lane);
        wv_fdyn(Xs, k, A1f, A2f, A3f, a1p, a2p, a3p, lane);            // k2
#pragma unroll
        for (int t = 0; t < 8; ++t)
#pragma unroll
            for (int p = 0; p < 8; ++p) sum[t][p] += 2.0f * k[t][p];
        wv_store_axpy(Xs, y, k, 0.5f * dt, lane);
        wv_fdyn(Xs, k, A1f, A2f, A3f, a1p, a2p, a3p, lane);            // k3
#pragma unroll
        for (int t = 0; t < 8; ++t)
#pragma unroll
            for (int p = 0; p < 8; ++p) sum[t][p] += 2.0f * k[t][p];
        wv_store_axpy(Xs, y, k, dt, lane);
        wv_fdyn(Xs, k, A1f, A2f, A3f, a1p, a2p, a3p, lane);            // k4
#pragma unroll
        for (int t = 0; t < 8; ++t)
#pragma unroll
            for (int p = 0; p < 8; ++p)
                y[t][p] += (dt / 6.0f) * (sum[t][p] + k[t][p]);
    }

    float oy[8];
    wv_ln_dot(y, g2, beta2, W2, b2s, oy, lane);

    if ((lane & 15) == 0) {
        const int half = lane >> 4;
#pragma unroll
        for (int p = 0; p < 8; ++p)
            out[base + p + 8 * half] = 0.5f * (oh[p] + oy[p]);
    }
}

// ---------------- kernel 2: sequential autoregressive tail (1 row) ----------------

__device__ __forceinline__ float tail_breduce(float v, float* red) {
    int t = threadIdx.x;
    red[t] = v;
    __syncthreads();
    for (int s = 64; s > 0; s >>= 1) {
        if (t < s) red[t] += red[t + s];
        __syncthreads();
    }
    float r = red[0];
    __syncthreads();
    return r;
}

__device__ void tail_fdyn(const float* __restrict__ yin, float* __restrict__ kout,
                          float* __restrict__ t1, float* __restrict__ t2,
                          const float* __restrict__ A1, const float* __restrict__ a1,
                          const float* __restrict__ A2, const float* __restrict__ a2,
                          const float* __restrict__ A3, const float* __restrict__ a3) {
    int t = threadIdx.x;
    __syncthreads();
    float a = 0.0f;
    if (t < 124) {
#pragma unroll 8
        for (int k = 0; k < 128; ++k) a += yin[k] * A1[k * 124 + t];
        a = fmaxf(a + a1[t], 0.0f);
    }
    t1[t] = (t < 124) ? a : 0.0f;
    __syncthreads();
    float b = 0.0f;
    if (t < 124) {
#pragma unroll 8
        for (int k = 0; k < 124; ++k) b += t1[k] * A2[k * 124 + t];
        b = fmaxf(b + a2[t], 0.0f);
    }
    t2[t] = (t < 124) ? b : 0.0f;
    __syncthreads();
    float c = 0.0f;
#pragma unroll 8
    for (int k = 0; k < 124; ++k) c += t2[k] * A3[k * 128 + t];
    kout[t] = tanh_fast(c + a3[t]);
    __syncthreads();
}

__device__ float tail_pipeline(const float* __restrict__ inp,
                               float* yv, float* yt, float* kv,
                               float* t1, float* t2, float* red,
                               const float* W1, const float* b1,
                               const float* g1, const float* beta1,
                               const float* A1, const float* a1,
                               const float* A2, const float* a2,
                               const float* A3, const float* a3,
                               const float* g2, const float* beta2,
                               const float* W2, float b2s) {
    int t = threadIdx.x;
    const float dt = 1.0f / (float)NSTEP;

    float acc = 0.0f;
#pragma unroll 8
    for (int k = 0; k < 128; ++k) acc += inp[k] * W1[k * 128 + t];
    acc = fmaxf(acc + b1[t], 0.0f);

    float m = tail_breduce(acc, red) * (1.0f / 128.0f);
    float d0 = acc - m;
    float var = tail_breduce(d0 * d0, red) * (1.0f / 128.0f);
    float hn = d0 * rsqrtf(var + LN_EPS) * g1[t] + beta1[t];

    float m2 = tail_breduce(hn, red) * (1.0f / 128.0f);
    float d2 = hn - m2;
    float v2 = tail_breduce(d2 * d2, red) * (1.0f / 128.0f);
    float dh = tail_breduce((d2 * rsqrtf(v2 + LN_EPS) * g2[t] + beta2[t]) * W2[t], red);

    yv[t] = hn;
    __syncthreads();

    for (int s = 0; s < NSTEP; ++s) {
        tail_fdyn(yv, kv, t1, t2, A1, a1, A2, a2, A3, a3);
        float sacc = kv[t];
        yt[t] = yv[t] + 0.5f * dt * kv[t];
        tail_fdyn(yt, kv, t1, t2, A1, a1, A2, a2, A3, a3);
        sacc += 2.0f * kv[t];
        yt[t] = yv[t] + 0.5f * dt * kv[t];
        tail_fdyn(yt, kv, t1, t2, A1, a1, A2, a2, A3, a3);
        sacc += 2.0f * kv[t];
        yt[t] = yv[t] + dt * kv[t];
        tail_fdyn(yt, kv, t1, t2, A1, a1, A2, a2, A3, a3);
        float yn = yv[t] + (dt / 6.0f) * (sacc + kv[t]);
        __syncthreads();
        yv[t] = yn;
        __syncthreads();
    }

    float yf = yv[t];
    float m3 = tail_breduce(yf, red) * (1.0f / 128.0f);
    float d3 = yf - m3;
    float v3 = tail_breduce(d3 * d3, red) * (1.0f / 128.0f);
    float dy = tail_breduce((d3 * rsqrtf(v3 + LN_EPS) * g2[t] + beta2[t]) * W2[t], red);

    return 0.5f * (dh + dy) + b2s;
}

__global__ __launch_bounds__(128) void ode_tail_kernel(
    const float* __restrict__ x,
    const float* __restrict__ W1, const float* __restrict__ b1,
    const float* __restrict__ g1, const float* __restrict__ beta1,
    const float* __restrict__ A1, const float* __restrict__ a1,
    const float* __restrict__ A2, const float* __restrict__ a2,
    const float* __restrict__ A3, const float* __restrict__ a3,
    const float* __restrict__ g2, const float* __restrict__ beta2,
    const float* __restrict__ W2, const float* __restrict__ b2,
    float* __restrict__ out) {
    __shared__ float lag[128], yv[128], yt[128], kv[128], t1[128], t2[128], red[128];
    __shared__ float prev;
    int t = threadIdx.x;
    const float b2s = b2[0];

    lag[t] = x[(size_t)(NB - 1) * DIM + t];
    if (t == 0) prev = out[NB - 1];
    __syncthreads();

    for (int i = 0; i < FUTN; ++i) {
        float nv = (t == 0) ? prev : lag[t - 1];
        __syncthreads();
        lag[t] = nv;
        __syncthreads();
        float res = tail_pipeline(lag, yv, yt, kv, t1, t2, red,
                                  W1, b1, g1, beta1, A1, a1, A2, a2, A3, a3,
                                  g2, beta2, W2, b2s);
        if (t == 0) {
            out[NB + i] = res;
            prev = res;
        }
        __syncthreads();
    }
}

// ---------------- launcher ----------------

extern "C" void kernel_launch(void* const* d_in, const int* in_sizes, int n_in,
                              void* d_out, int out_size, void* d_ws, size_t ws_size,
                              hipStream_t stream) {
    const float* x     = (const float*)d_in[0];
    const float* W1    = (const float*)d_in[1];
    const float* b1    = (const float*)d_in[2];
    const float* g1    = (const float*)d_in[3];
    const float* beta1 = (const float*)d_in[4];
    const float* A1    = (const float*)d_in[5];
    const float* a1    = (const float*)d_in[6];
    const float* A2    = (const float*)d_in[7];
    const float* a2    = (const float*)d_in[8];
    const float* A3    = (const float*)d_in[9];
    const float* a3    = (const float*)d_in[10];
    const float* g2    = (const float*)d_in[11];
    const float* beta2 = (const float*)d_in[12];
    const float* W2    = (const float*)d_in[13];
    const float* b2    = (const float*)d_in[14];
    float* out = (float*)d_out;
    float* ws  = (float*)d_ws;

    const float* W1f = ws;
    const float* A1f = ws + 16384;
    const float* A2f = ws + 32768;
    const float* A3f = ws + 49152;
    const float* b1p = ws + 65536;
    const float* a1p = ws + 65664;
    const float* a2p = ws + 65792;
    const float* a3p = ws + 65920;

    ode_prep_kernel<<<(66048 + 255) / 256, 256, 0, stream>>>(
        W1, A1, A2, A3, b1, a1, a2, a3, ws);

    ode_batch_kernel<<<NB / 16 / 4, 128, 0, stream>>>(
        x, g1, beta1, g2, beta2, W2, b2,
        W1f, A1f, A2f, A3f, b1p, a1p, a2p, a3p, out);

    ode_tail_kernel<<<1, 128, 0, stream>>>(
        x, W1, b1, g1, beta1, A1, a1, A2, a2, A3, a3, g2, beta2, W2, b2, out);
}